// ResonantLayer_11965778886782
// MI455X (gfx1250) — compile-verified
//
#include <hip/hip_runtime.h>
#include <math.h>

typedef float v2f __attribute__((ext_vector_type(2)));
typedef float v8f __attribute__((ext_vector_type(8)));

#define B_SZ   128
#define D_SZ   1024
#define N_SZ   1024
#define K_COL  2048          // collapse GEMM K (2*D)
#define LUTN   4096

// ---------------------------------------------------------------------------
// Kernel 0: build interleaved (sin,cos) LUT, padded with a wrap entry.
// Entry i = (sin(i*2pi/4096), cos(i*2pi/4096)) in fp32, entry 4096 = entry 0
// (matches reference's mod(i0+1, 4096) exactly).
// ---------------------------------------------------------------------------
__global__ void build_lut_kernel(float2* __restrict__ lut) {
    int i = blockIdx.x * blockDim.x + threadIdx.x;
    if (i < LUTN) {
        const float step = 6.28318530717958647692f / (float)LUTN;
        float th = (float)i * step;
        float2 e; e.x = sinf(th); e.y = cosf(th);
        lut[i] = e;
        if (i == 0) lut[LUTN] = e;   // wrap pad == entry 0
    }
}

// ---------------------------------------------------------------------------
// Kernel 1: x = concat(x_real, x_imag) @ Wc + bc   (128x2048 * 2048x1024)
// One wave per 16x16 output tile, fp32 WMMA 16x16x4.
// ---------------------------------------------------------------------------
__global__ void collapse_gemm_kernel(const float* __restrict__ xr,
                                     const float* __restrict__ xi,
                                     const float* __restrict__ Wc,
                                     const float* __restrict__ bc,
                                     float* __restrict__ xout) {
    const int gwave = (blockIdx.x * blockDim.x + threadIdx.x) >> 5;
    const int lane  = threadIdx.x & 31;
    const int half  = lane >> 4;        // 0: K pair {0,1}, 1: K pair {2,3}
    const int l     = lane & 15;
    const int mt = gwave >> 6;          // 8 M-tiles
    const int nt = gwave & 63;          // 64 N-tiles
    const int m0 = mt * 16, n0 = nt * 16;
    const int row = m0 + l;             // A row this lane loads
    const int col = n0 + l;             // B col this lane loads

    v8f c = {};
    for (int k0 = 0; k0 < K_COL; k0 += 4) {
        const int ka = k0 + 2 * half;
        // A = [x_real | x_imag]; ka and ka+1 are on the same side of the split
        const float* asrc = (ka < D_SZ) ? (xr + row * D_SZ + ka)
                                        : (xi + row * D_SZ + (ka - D_SZ));
        v2f a; a.x = asrc[0]; a.y = asrc[1];
        v2f b; b.x = Wc[(size_t)ka * D_SZ + col];
               b.y = Wc[(size_t)(ka + 1) * D_SZ + col];
        c = __builtin_amdgcn_wmma_f32_16x16x4_f32(
                false, a, false, b, (short)0, c, false, false);
    }
    const float bias = bc[col];
    #pragma unroll
    for (int v = 0; v < 8; ++v)
        xout[(size_t)(m0 + v + 8 * half) * D_SZ + col] = c[v] + bias;
}

// ---------------------------------------------------------------------------
// Kernel 2: resonance core. For each (b,n):
//   theta_d = x[b,d]/(1+|Wfreq[n,d]|) + Bias[n,d] + t[b]
//   LUT-interp sin/cos, reduce over d against attn_sin/attn_cos.
// Interleaved padded LUT (32KB) + x-row (4KB) in LDS.
// Each lane processes 4 consecutive d -> b128 global loads, b64 LDS gathers.
// ---------------------------------------------------------------------------
__global__ void __launch_bounds__(256)
resonance_kernel(const float* __restrict__ x,
                 const float* __restrict__ t,
                 const float* __restrict__ Wfreq,
                 const float* __restrict__ Bias,
                 const float* __restrict__ attn_cos,
                 const float* __restrict__ attn_sin,
                 const float2* __restrict__ lut,
                 float* __restrict__ cos_sum,
                 float* __restrict__ sin_sum) {
    __shared__ float2 s_lut[LUTN + 1];
    __shared__ float  s_x[D_SZ];

    const int tid = threadIdx.x;
    const int b   = blockIdx.x;

    for (int i = tid; i < LUTN + 1; i += 256)
        s_lut[i] = lut[i];
    for (int i = tid; i < D_SZ; i += 256)
        s_x[i] = x[(size_t)b * D_SZ + i];
    __syncthreads();

    const float tb    = t[b];
    const int   lane  = tid & 31;
    const int   wave  = tid >> 5;          // 8 waves per block
    const int   nbase = blockIdx.y * 32;   // 32 n-values per block
    const float scale = 651.89864690440329f; // 4096 / (2*pi), f32

    for (int j = 0; j < 4; ++j) {
        const int n = nbase + wave * 4 + j;
        const size_t rowoff = (size_t)n * D_SZ;
        const float* wf = Wfreq    + rowoff;
        const float* bs = Bias     + rowoff;
        const float* cc = attn_cos + rowoff;
        const float* ss = attn_sin + rowoff;

        if (j < 3) {
            // prefetch next n-row into L2/L0 (global_prefetch_b8)
            const size_t nx = rowoff + D_SZ + (size_t)lane * 32;
            __builtin_prefetch(Wfreq    + nx, 0, 1);
            __builtin_prefetch(Bias     + nx, 0, 1);
            __builtin_prefetch(attn_cos + nx, 0, 1);
            __builtin_prefetch(attn_sin + nx, 0, 1);
        }

        float ac = 0.0f, as = 0.0f;
        #pragma unroll 2
        for (int i = 0; i < D_SZ / 128; ++i) {
            const int d = lane * 4 + i * 128;   // 4 consecutive d per lane
            const float4 wf4 = *(const float4*)(wf + d);
            const float4 bs4 = *(const float4*)(bs + d);
            const float4 cc4 = *(const float4*)(cc + d);
            const float4 ss4 = *(const float4*)(ss + d);
            const float4 sx4 = *(const float4*)(s_x + d);
            const float* wfp = (const float*)&wf4;
            const float* bsp = (const float*)&bs4;
            const float* ccp = (const float*)&cc4;
            const float* ssp = (const float*)&ss4;
            const float* sxp = (const float*)&sx4;
            #pragma unroll
            for (int q = 0; q < 4; ++q) {
                float wav   = 1.0f + fabsf(wfp[q]);
                float theta = sxp[q] / wav + bsp[q] + tb;
                float pos   = theta * scale;
                float i0f   = floorf(pos);
                float w     = pos - i0f;
                int   i0    = ((int)i0f) & (LUTN - 1); // == mod 4096
                float2 e0 = s_lut[i0];       // (sin,cos) at i0  : ds_load_b64
                float2 e1 = s_lut[i0 + 1];   // (sin,cos) at i0+1: padded, no wrap
                float st = (1.0f - w) * e0.x + w * e1.x;
                float ct = (1.0f - w) * e0.y + w * e1.y;
                ac += ct * ccp[q];
                as += st * ssp[q];
            }
        }
        // wave32 butterfly reduction
        #pragma unroll
        for (int off = 16; off > 0; off >>= 1) {
            ac += __shfl_xor(ac, off, 32);
            as += __shfl_xor(as, off, 32);
        }
        if (lane == 0) {
            cos_sum[(size_t)b * N_SZ + n] = ac;
            sin_sum[(size_t)b * N_SZ + n] = as;
        }
    }
}

// ---------------------------------------------------------------------------
// Kernel 3: out = silu(S @ W)   (128x1024 * 1024x1024), fp32 WMMA + fused silu
// ---------------------------------------------------------------------------
__global__ void proj_silu_gemm_kernel(const float* __restrict__ S,
                                      const float* __restrict__ W,
                                      float* __restrict__ out) {
    const int gwave = (blockIdx.x * blockDim.x + threadIdx.x) >> 5;
    const int lane  = threadIdx.x & 31;
    const int half  = lane >> 4;
    const int l     = lane & 15;
    const int mt = gwave >> 6;
    const int nt = gwave & 63;
    const int m0 = mt * 16, n0 = nt * 16;
    const int row = m0 + l;
    const int col = n0 + l;

    v8f c = {};
    for (int k0 = 0; k0 < N_SZ; k0 += 4) {
        const int ka = k0 + 2 * half;
        v2f a; a.x = S[(size_t)row * N_SZ + ka];
               a.y = S[(size_t)row * N_SZ + ka + 1];
        v2f b; b.x = W[(size_t)ka * D_SZ + col];
               b.y = W[(size_t)(ka + 1) * D_SZ + col];
        c = __builtin_amdgcn_wmma_f32_16x16x4_f32(
                false, a, false, b, (short)0, c, false, false);
    }
    #pragma unroll
    for (int v = 0; v < 8; ++v) {
        float z = c[v];
        float sig = 1.0f / (1.0f + __expf(-z));
        out[(size_t)(m0 + v + 8 * half) * D_SZ + col] = z * sig;
    }
}

// ---------------------------------------------------------------------------
extern "C" void kernel_launch(void* const* d_in, const int* in_sizes, int n_in,
                              void* d_out, int out_size, void* d_ws, size_t ws_size,
                              hipStream_t stream) {
    const float* xr       = (const float*)d_in[0];
    const float* xi       = (const float*)d_in[1];
    const float* t        = (const float*)d_in[2];
    const float* Wc       = (const float*)d_in[3];
    const float* bc       = (const float*)d_in[4];
    const float* Wfreq    = (const float*)d_in[5];
    const float* Bias     = (const float*)d_in[6];
    const float* attn_cos = (const float*)d_in[7];
    const float* attn_sin = (const float*)d_in[8];
    const float* Wr       = (const float*)d_in[9];
    const float* Wi       = (const float*)d_in[10];
    float* out = (float*)d_out;

    float*  ws   = (float*)d_ws;
    float2* lut  = (float2*)ws;                          // (4096+1) float2
    float*  xcol = ws + 2 * (LUTN + 2);                  // 128*1024 (8B-aligned past lut)
    float*  csum = xcol + (size_t)B_SZ * D_SZ;           // 128*1024
    float*  ssum = csum + (size_t)B_SZ * N_SZ;           // 128*1024

    build_lut_kernel<<<LUTN / 256, 256, 0, stream>>>(lut);

    // 512 tiles, 8 waves (256 threads) per block -> 64 blocks
    collapse_gemm_kernel<<<64, 256, 0, stream>>>(xr, xi, Wc, bc, xcol);

    dim3 rg(B_SZ, N_SZ / 32);
    resonance_kernel<<<rg, 256, 0, stream>>>(xcol, t, Wfreq, Bias,
                                             attn_cos, attn_sin,
                                             lut, csum, ssum);

    proj_silu_gemm_kernel<<<64, 256, 0, stream>>>(csum, Wr, out);
    proj_silu_gemm_kernel<<<64, 256, 0, stream>>>(ssum, Wi, out + (size_t)B_SZ * D_SZ);
}